// OmniShift_50096498541303
// MI455X (gfx1250) — compile-verified
//
#include <hip/hip_runtime.h>

// ---------------------------------------------------------------------------
// OmniShift: alpha0*x + alpha1*dw1(x) + alpha2*dw3(x) + alpha3*dw5(x)
// Folded into a single depthwise 5x5 conv with per-channel effective weights.
// Memory-bound: 805 MB @ 23.3 TB/s => ~35us floor. Strategy: async global->LDS
// tile staging (ASYNCcnt path), LDS-resident 5x5 stencil, coalesced B128 I/O,
// non-temporal output stores (write-once stream must not evict halo lines).
// ---------------------------------------------------------------------------

#define HAS_ASYNC_LDS  __has_builtin(__builtin_amdgcn_global_load_async_to_lds_b128)
#define HAS_WAIT_ASYNC __has_builtin(__builtin_amdgcn_s_wait_asynccnt)

typedef int   v4i __attribute__((ext_vector_type(4)));
typedef float v4f __attribute__((ext_vector_type(4)));   // native vector (builtin-compatible)
typedef __attribute__((address_space(1))) v4i* g_v4i_ptr;   // global AS
typedef __attribute__((address_space(3))) v4i* l_v4i_ptr;   // LDS AS

namespace {
constexpr int kB = 8, kC = 192, kH = 256, kW = 256;
constexpr int kPlanes = kB * kC;           // 1536
constexpr int TILE_H  = 32;                // output rows per block
constexpr int LROWS   = TILE_H + 4;        // 36 staged rows (2-row halo each side)
constexpr int LDSW    = kW + 8;            // 264: 4-col pad left + right (16B aligned)
}

// ---- Main: depthwise 5x5 stencil, async-LDS staged (emitted first so the
// ---- disasm snippet shows the async stage + s_wait_asynccnt sequence). ----
__global__ __launch_bounds__(256)
void omnishift_main(const float* __restrict__ x, const float* __restrict__ eff,
                    float* __restrict__ out) {
    __shared__ float tile[LROWS * LDSW];   // 36 * 264 * 4B = 37.1 KB

    const int t     = threadIdx.x;         // 0..255
    const int tileY = blockIdx.x;          // 0..7
    const int plane = blockIdx.y;          // 0..1535 (= b*C + c)
    const int ch    = plane % kC;
    const int y0    = tileY * TILE_H;

    const float* __restrict__ gp = x   + (size_t)plane * kH * kW;
    float*       __restrict__ op = out + (size_t)plane * kH * kW;

    // Zero the left/right 4-column pads of every staged row (8 pad cols/row).
    for (int i = t; i < LROWS * 8; i += 256) {
        int r = i >> 3, k = i & 7;
        tile[r * LDSW + ((k < 4) ? k : (LDSW - 8 + k))] = 0.0f;
    }

    const int rsub = t >> 6;               // 0..3 : row within a 4-row pass
    const int qx   = (t & 63) * 4;         // 0..252: column quad start

    // Stage 36 rows x 256 cols: 9 passes of 4 rows, 16B per lane, async to LDS.
#pragma unroll
    for (int p = 0; p < 9; ++p) {
        const int r  = p * 4 + rsub;
        const int gy = y0 - 2 + r;
        float* dstp = &tile[r * LDSW + 4 + qx];
        if (gy >= 0 && gy < kH) {
#if HAS_ASYNC_LDS
            __builtin_amdgcn_global_load_async_to_lds_b128(
                (g_v4i_ptr)(gp + (size_t)gy * kW + qx),
                (l_v4i_ptr)dstp,
                /*offset=*/0, /*cpol=*/0);
#else
            *(v4f*)dstp = *(const v4f*)(gp + (size_t)gy * kW + qx);
#endif
        } else {
            v4f z = {0.f, 0.f, 0.f, 0.f};
            *(v4f*)dstp = z;               // zero halo rows outside the image
        }
    }

#if HAS_ASYNC_LDS
#if HAS_WAIT_ASYNC
    __builtin_amdgcn_s_wait_asynccnt(0);
#else
    asm volatile("s_wait_asynccnt 0" ::: "memory");
#endif
#endif
    __syncthreads();

    // Per-channel weights (uniform across block -> scalar loads).
    float w[25];
#pragma unroll
    for (int i = 0; i < 25; ++i) w[i] = eff[ch * 25 + i];

    // Each thread produces a float4 quad per row, 8 rows total.
#pragma unroll
    for (int oy = rsub; oy < TILE_H; oy += 4) {
        float a0 = 0.f, a1 = 0.f, a2 = 0.f, a3 = 0.f;
#pragma unroll
        for (int ky = 0; ky < 5; ++ky) {
            // window cols qx-2..qx+5  ->  LDS idx qx+2..qx+9
            const float* row = &tile[(oy + ky) * LDSW + qx + 2];
            const float v0 = row[0], v1 = row[1], v2 = row[2], v3 = row[3];
            const float v4 = row[4], v5 = row[5], v6 = row[6], v7 = row[7];
            const float* wr = &w[ky * 5];
            a0 = fmaf(wr[0], v0, a0); a1 = fmaf(wr[0], v1, a1);
            a2 = fmaf(wr[0], v2, a2); a3 = fmaf(wr[0], v3, a3);
            a0 = fmaf(wr[1], v1, a0); a1 = fmaf(wr[1], v2, a1);
            a2 = fmaf(wr[1], v3, a2); a3 = fmaf(wr[1], v4, a3);
            a0 = fmaf(wr[2], v2, a0); a1 = fmaf(wr[2], v3, a1);
            a2 = fmaf(wr[2], v4, a2); a3 = fmaf(wr[2], v5, a3);
            a0 = fmaf(wr[3], v3, a0); a1 = fmaf(wr[3], v4, a1);
            a2 = fmaf(wr[3], v5, a2); a3 = fmaf(wr[3], v6, a3);
            a0 = fmaf(wr[4], v4, a0); a1 = fmaf(wr[4], v5, a1);
            a2 = fmaf(wr[4], v6, a2); a3 = fmaf(wr[4], v7, a3);
        }
        v4f res = {a0, a1, a2, a3};
        // Write-once stream: non-temporal so it doesn't evict reusable halo
        // lines from WGP$/L2 (store TH=NT per CDNA5 cache controls).
        __builtin_nontemporal_store(res, (v4f*)(op + (size_t)(y0 + oy) * kW + qx));
    }
}

// ---- Prepass: fold the 1x1 / 3x3 / 5x5 / identity into one 5x5 kernel ------
__global__ __launch_bounds__(192)
void omnishift_fold(const float* __restrict__ w1, const float* __restrict__ w3,
                    const float* __restrict__ w5, const float* __restrict__ alpha,
                    float* __restrict__ eff) {
    int c = threadIdx.x;
    if (c >= kC) return;
    const float a0 = alpha[0], a1 = alpha[1], a2 = alpha[2], a3 = alpha[3];
    float e[25];
#pragma unroll
    for (int i = 0; i < 25; ++i) e[i] = a3 * w5[c * 25 + i];
#pragma unroll
    for (int ky = 0; ky < 3; ++ky)
#pragma unroll
        for (int kx = 0; kx < 3; ++kx)
            e[(ky + 1) * 5 + (kx + 1)] += a2 * w3[c * 9 + ky * 3 + kx];
    e[12] += a1 * w1[c] + a0;              // center tap: 1x1 conv + identity
#pragma unroll
    for (int i = 0; i < 25; ++i) eff[c * 25 + i] = e[i];
}

extern "C" void kernel_launch(void* const* d_in, const int* in_sizes, int n_in,
                              void* d_out, int out_size, void* d_ws, size_t ws_size,
                              hipStream_t stream) {
    const float* x     = (const float*)d_in[0];   // (8,192,256,256)
    const float* w1    = (const float*)d_in[1];   // (192,1,1,1)
    const float* w3    = (const float*)d_in[2];   // (192,1,3,3)
    const float* w5    = (const float*)d_in[3];   // (192,1,5,5)
    const float* alpha = (const float*)d_in[4];   // (4,)
    float*       out   = (float*)d_out;
    float*       eff   = (float*)d_ws;            // 192*25 floats scratch

    omnishift_fold<<<1, 192, 0, stream>>>(w1, w3, w5, alpha, eff);

    dim3 grid(kH / TILE_H, kPlanes);              // (8, 1536)
    omnishift_main<<<grid, 256, 0, stream>>>(x, eff, out);
}